// LearnedPositionalEmbedding_19894288515422
// MI455X (gfx1250) — compile-verified
//
#include <hip/hip_runtime.h>
#include <cstdint>
#include <cstddef>

// Problem constants (from the reference)
#define PE_MAX_LENGTH 8192
#define PE_MODEL_DIM  1024
#define PE_BATCH      4
#define PE_SEQ        4096

// Tiling: 32x32 transpose tile, 256 threads (8 wave32s) per workgroup.
#define TILE       32
#define LDS_STRIDE 33   // odd stride -> conflict-free transposed LDS reads (33 coprime 64)

#if defined(__AMDGCN__) && __has_builtin(__builtin_amdgcn_global_load_async_to_lds_b32)
#define PE_ASYNC_BUILTIN 1
#else
#define PE_ASYNC_BUILTIN 0
#endif

#if defined(__gfx1250__)
#define PE_GFX1250 1
#else
#define PE_GFX1250 0
#endif

typedef __attribute__((address_space(1))) int GlobalInt;   // prints as '__device__ int *'
typedef __attribute__((address_space(3))) int LdsInt;      // LDS / '__shared__'
typedef __attribute__((address_space(3))) void LdsVoid;

// out[b, s, d] = W[d, s] + bias[d]   for b in 0..3
// W is [MODEL_DIM, MAX_LENGTH] row-major -> gathering column s is a transpose.
__global__ __launch_bounds__(256) void pe_transpose_bcast_kernel(
    const float* __restrict__ W,
    const float* __restrict__ bias,
    float* __restrict__ out)
{
    __shared__ float tile[TILE * LDS_STRIDE];

    const int tx = threadIdx.x;            // 0..31 (one wave lane span)
    const int ty = threadIdx.y;            // 0..7
    const int s0 = blockIdx.x * TILE;      // sequence-tile origin
    const int d0 = blockIdx.y * TILE;      // model-dim-tile origin

    // ---- Stage W tile (rows = d, cols = s) into LDS, coalesced along s ----
#if PE_ASYNC_BUILTIN
    #pragma unroll
    for (int j = 0; j < 4; ++j) {
        const int dr = ty + j * 8;
        const float* gsrc = W + (size_t)(d0 + dr) * PE_MAX_LENGTH + (s0 + tx);
        float* lds = &tile[dr * LDS_STRIDE + tx];
        // async global -> LDS copy (ASYNCcnt path), 4B per lane, 128B per wave
        __builtin_amdgcn_global_load_async_to_lds_b32(
            (GlobalInt*)gsrc, (LdsInt*)lds, /*offset=*/0, /*cpol=*/0);
    }
#if __has_builtin(__builtin_amdgcn_s_wait_asynccnt)
    __builtin_amdgcn_s_wait_asynccnt(0);
#else
    asm volatile("s_wait_asynccnt 0" ::: "memory");
#endif
#elif PE_GFX1250
    #pragma unroll
    for (int j = 0; j < 4; ++j) {
        const int dr = ty + j * 8;
        const uint64_t gaddr =
            (uint64_t)(uintptr_t)(W + (size_t)(d0 + dr) * PE_MAX_LENGTH + (s0 + tx));
        const uint32_t laddr =
            (uint32_t)(__UINTPTR_TYPE__)(LdsVoid*)&tile[dr * LDS_STRIDE + tx];
        // GV form: vdst = LDS byte address, vaddr = 64-bit global address
        asm volatile("global_load_async_to_lds_b32 %0, %1, off"
                     :: "v"(laddr), "v"(gaddr) : "memory");
    }
    asm volatile("s_wait_asynccnt 0" ::: "memory");
#else
    #pragma unroll
    for (int j = 0; j < 4; ++j) {
        const int dr = ty + j * 8;
        tile[dr * LDS_STRIDE + tx] = W[(size_t)(d0 + dr) * PE_MAX_LENGTH + (s0 + tx)];
    }
#endif
    __syncthreads();

    // ---- Write phase: coalesced along d, broadcast to all 4 batches ----
    const float bv = bias[d0 + tx];

    #pragma unroll
    for (int j = 0; j < 4; ++j) {
        const int sr = ty + j * 8;
        const float v = tile[tx * LDS_STRIDE + sr] + bv;   // transposed LDS read
        const size_t base = (size_t)(s0 + sr) * PE_MODEL_DIM + (size_t)(d0 + tx);
        #pragma unroll
        for (int bb = 0; bb < PE_BATCH; ++bb) {
            out[(size_t)bb * (size_t)PE_SEQ * PE_MODEL_DIM + base] = v;
        }
    }
}

extern "C" void kernel_launch(void* const* d_in, const int* in_sizes, int n_in,
                              void* d_out, int out_size, void* d_ws, size_t ws_size,
                              hipStream_t stream) {
    (void)in_sizes; (void)n_in; (void)out_size; (void)d_ws; (void)ws_size;
    // Input order from setup_inputs(): x (unused), W, b
    const float* W = (const float*)d_in[1];
    const float* b = (const float*)d_in[2];
    float* out     = (float*)d_out;

    dim3 grid(PE_SEQ / TILE, PE_MODEL_DIM / TILE);   // 128 x 32 tiles
    dim3 block(TILE, 8);                             // 256 threads = 8 wave32s
    pe_transpose_bcast_kernel<<<grid, block, 0, stream>>>(W, b, out);
}